// MILRouting_54434415509639
// MI455X (gfx1250) — compile-verified
//
#include <hip/hip_runtime.h>
#include <stdint.h>

// ---------------------------------------------------------------------------
// MIL dynamic routing on MI455X (gfx1250).
//
// weights_k = nodes * Scum_k[indices]  with  Scum_k = sum_{j<k} sigma_j,
// so we keep only a 4MB L2-resident cumulative-sigma table instead of the
// 205MB weights tensor. Memory-bound: 5 streams of nodes (~1GB) ~= 44us at
// 23.3 TB/s. nodes tiles are staged into LDS via the Tensor Data Mover
// (tensor_load_to_lds, double buffered, s_wait_tensorcnt). Segment scatter
// uses native global_atomic_add_f32 (forced via inline asm so no CAS-loop
// fallback is possible).
// ---------------------------------------------------------------------------

#define N_NODES   200000
#define DIMS      256
#define NBAGS     4096
#define TR        16                    // tile rows (200000 % 16 == 0)
#define TILES     (N_NODES / TR)        // 12500
#define GRID_STREAM 1250                // 12500 / 1250 = exactly 10 tiles/block

typedef unsigned int u32;
typedef unsigned long long u64;
typedef __attribute__((ext_vector_type(4))) u32 v4u;
typedef __attribute__((ext_vector_type(4))) int v4i;
typedef __attribute__((ext_vector_type(8))) int v8i;

// Order-preserving float <-> uint encoding so atomicMax(u32) == float max.
__device__ __forceinline__ u32 encf(float f) {
  u32 b = __float_as_uint(f);
  return (b & 0x80000000u) ? ~b : (b | 0x80000000u);
}
__device__ __forceinline__ float decf(u32 u) {
  return __uint_as_float((u & 0x80000000u) ? (u ^ 0x80000000u) : ~u);
}
#define ENC_NEG_INF 0x007FFFFFu   // encf(-inf)
#define ENC_ZERO    0x80000000u   // encf(0.0f)

// Native fp32 global atomic add, no return (STOREcnt path, device scope).
// Inline asm guarantees the hardware instruction -- a compiler CAS-loop
// expansion would be ~10x slower on 51M adds/pass.
__device__ __forceinline__ void fatomic_add(float* p, float v) {
  asm volatile("global_atomic_add_f32 %0, %1, off scope:SCOPE_DEV"
               :
               : "v"((u64)(uintptr_t)p), "v"(v)
               : "memory");
}

// Issue one TDM DMA: copy a TR x 256 fp32 tile (contiguous rows, row stride
// 256) from global memory into LDS at byte offset lds_off. Descriptor built
// per CDNA5 ISA (group0 + group1; groups 2/3 and the
// trailing group are zero since tile_dim2/3 == 0 for a 2D tile).
// Tensor dims == tile dims -> exact copy, no OOB handling needed.
__device__ __forceinline__ void tdm_load_tile(const float* gsrc, u32 lds_off) {
  u64 ga = (u64)(uintptr_t)gsrc;
  v4u g0;
  g0.x = 1u;                                        // count=1, user mode
  g0.y = lds_off;                                   // lds_addr
  g0.z = (u32)ga;                                   // global_addr[31:0]
  g0.w = ((u32)(ga >> 32) & 0x01FFFFFFu) | (2u << 30); // addr[56:32] | type=2
  v8i g1;
  g1[0] = 0x00020000;              // workgroup_mask=0, data_size=2 (4 bytes)
  g1[1] = (int)(DIMS << 16);       // tensor_dim0 = 256 (bits 79:48 low half)
  g1[2] = (int)(TR << 16);         // tensor_dim0 hi = 0 | tensor_dim1 = TR
  g1[3] = (int)(DIMS << 16);       // tensor_dim1 hi = 0 | tile_dim0 = 256
  g1[4] = TR;                      // tile_dim1 = TR, tile_dim2 = 0
  g1[5] = DIMS;                    // tensor_dim0_stride = 256 (low 32)
  g1[6] = 0;                       // stride hi, tensor_dim1_stride lo
  g1[7] = 0;                       // tensor_dim1_stride hi
  v4i gz4 = {0, 0, 0, 0};
  v8i gz8 = {0, 0, 0, 0, 0, 0, 0, 0};
  // amdgpu-toolchain (clang-23) 6-arg form: (g0, g1, g2, g3, extra, cpol)
  __builtin_amdgcn_tensor_load_to_lds(g0, g1, gz4, gz4, gz8, 0);
  asm volatile("" ::: "memory");
}

// One streaming pass over nodes, TDM double-buffered through LDS.
// MODE 0: per-column max of w = node * Scum[bag]   -> colMaxEnc (atomic max)
// MODE 1: e = exp(w - colMax); colZ += e; sigU[bag] += e * node (scatter)
template <int MODE>
__global__ __launch_bounds__(256) void stream_pass(
    const float* __restrict__ nodes, const int* __restrict__ idx,
    const float* __restrict__ Scum, u32* __restrict__ colMaxEnc,
    float* __restrict__ colZ, float* __restrict__ sigU) {
  __shared__ float tile[2][TR * DIMS];   // 32 KB
  const int d = threadIdx.x;             // thread == column
  const bool issuer = (threadIdx.x < 32);  // wave 0 drives the TDM
  const u32 lds0 = (u32)(uintptr_t)&tile[0][0];
  const u32 lds1 = (u32)(uintptr_t)&tile[1][0];

  float m = -INFINITY;
  float z = 0.0f;
  float mcol = 0.0f;
  if (MODE == 1) mcol = decf(colMaxEnc[d]);   // k=0: ENC_ZERO -> 0.0

  const int t0 = blockIdx.x;
  if (issuer) tdm_load_tile(nodes + (size_t)t0 * (TR * DIMS), lds0);

  int buf = 0;
  for (int t = t0; t < TILES; t += GRID_STREAM) {
    const int tn = t + GRID_STREAM;
    if (issuer) {
      if (tn < TILES) {   // prefetch next tile into the other buffer
        tdm_load_tile(nodes + (size_t)tn * (TR * DIMS), buf ? lds0 : lds1);
        __builtin_amdgcn_s_wait_tensorcnt(1);   // current tile done (in-order)
      } else {
        __builtin_amdgcn_s_wait_tensorcnt(0);
      }
    }
    __syncthreads();                      // tile[buf] visible to all waves
    const float* tl = &tile[buf][0];
    const int r0 = t * TR;
#pragma unroll
    for (int r = 0; r < TR; ++r) {
      const int b = idx[r0 + r];                       // uniform scalar load
      const float nv = tl[r * DIMS + d];               // LDS, conflict-free
      const float w = nv * Scum[b * DIMS + d];         // 4MB table: L2 hit
      if (MODE == 0) {
        m = fmaxf(m, w);
      } else {
        const float e = __expf(w - mcol);
        z += e;
        fatomic_add(&sigU[b * DIMS + d], e * nv);      // segment scatter-add
      }
    }
    __syncthreads();   // all reads of tile[buf] done before it is re-filled
    buf ^= 1;
  }

  if (MODE == 0) atomicMax(&colMaxEnc[d], encf(m));    // 256 atomics/block
  else           fatomic_add(&colZ[d], z);
}

__global__ void init_kernel(float* Scum, float* sigU, float* colZ,
                            u32* colMaxEnc) {
  const int i = blockIdx.x * blockDim.x + threadIdx.x;
  if (i < NBAGS * DIMS) { Scum[i] = 0.0f; sigU[i] = 0.0f; }
  if (blockIdx.x == 0 && threadIdx.x < DIMS) {
    colZ[threadIdx.x] = 0.0f;
    colMaxEnc[threadIdx.x] = ENC_ZERO;   // iter 0: w==0 everywhere -> max=0
  }
}

// sigma = sigU / Z; accumulate into Scum (or emit final output); clear sigU.
template <int LAST>
__global__ void normalize_kernel(float* __restrict__ Scum,
                                 float* __restrict__ sigU,
                                 const float* __restrict__ colZ,
                                 float* __restrict__ out) {
  const int i = blockIdx.x * blockDim.x + threadIdx.x;
  if (i < NBAGS * DIMS) {
    const float s = sigU[i] / colZ[i & (DIMS - 1)];
    if (LAST) {
      out[i] = s;
    } else {
      Scum[i] += s;
      sigU[i] = 0.0f;
    }
  }
}

__global__ void reset_stats(float* colZ, u32* colMaxEnc) {
  if (threadIdx.x < DIMS) {
    colZ[threadIdx.x] = 0.0f;
    colMaxEnc[threadIdx.x] = ENC_NEG_INF;
  }
}

extern "C" void kernel_launch(void* const* d_in, const int* in_sizes, int n_in,
                              void* d_out, int out_size, void* d_ws,
                              size_t ws_size, hipStream_t stream) {
  const float* nodes = (const float*)d_in[0];
  const int* idx = (const int*)d_in[1];
  float* out = (float*)d_out;

  // workspace layout: Scum[1M] | sigU[1M] | colZ[256] | colMaxEnc[256]
  float* Scum = (float*)d_ws;
  float* sigU = Scum + NBAGS * DIMS;
  float* colZ = sigU + NBAGS * DIMS;
  u32* colMaxEnc = (u32*)(colZ + DIMS);

  const dim3 blk(256);
  const dim3 gridNB((NBAGS * DIMS + 255) / 256);
  const dim3 gridS(GRID_STREAM);

  init_kernel<<<gridNB, blk, 0, stream>>>(Scum, sigU, colZ, colMaxEnc);

  // k = 0: softmax over zero [N,1] weights is uniform; Scum=0, max=0 gives
  // e = exp(0) = 1, Z = N  ->  sigma1 = segment_sum(nodes) / N.
  stream_pass<1><<<gridS, blk, 0, stream>>>(nodes, idx, Scum, colMaxEnc, colZ, sigU);
  normalize_kernel<0><<<gridNB, blk, 0, stream>>>(Scum, sigU, colZ, out);
  reset_stats<<<1, blk, 0, stream>>>(colZ, colMaxEnc);

  // k = 1
  stream_pass<0><<<gridS, blk, 0, stream>>>(nodes, idx, Scum, colMaxEnc, colZ, sigU);
  stream_pass<1><<<gridS, blk, 0, stream>>>(nodes, idx, Scum, colMaxEnc, colZ, sigU);
  normalize_kernel<0><<<gridNB, blk, 0, stream>>>(Scum, sigU, colZ, out);
  reset_stats<<<1, blk, 0, stream>>>(colZ, colMaxEnc);

  // k = 2 (final sigma -> d_out)
  stream_pass<0><<<gridS, blk, 0, stream>>>(nodes, idx, Scum, colMaxEnc, colZ, sigU);
  stream_pass<1><<<gridS, blk, 0, stream>>>(nodes, idx, Scum, colMaxEnc, colZ, sigU);
  normalize_kernel<1><<<gridNB, blk, 0, stream>>>(Scum, sigU, colZ, out);
}